// Model_70411693850654
// MI455X (gfx1250) — compile-verified
//
#include <hip/hip_runtime.h>

#define DEVINL __device__ __forceinline__

typedef float v2f __attribute__((ext_vector_type(2)));
typedef float v8f __attribute__((ext_vector_type(8)));

// ---------------- real Wigner-3j table offsets (flattened i*(2l2+1)*(2l3+1)+j*(2l3+1)+k) ----
#define OFF110 0
#define OFF111 9
#define OFF112 36
#define OFF121 81
#define OFF122 126
#define OFF211 201
#define OFF212 246
#define OFF220 321
#define OFF221 346
#define OFF222 421
#define OFF312 546
#define OFF321 651
#define OFF322 756
#define C3J_TOTAL 931

DEVINL double dfact(int n){ double r=1.0; for(int i=2;i<=n;++i) r*=(double)i; return r; }
DEVINL int imaxi(int a,int b){return a>b?a:b;}
DEVINL int imini(int a,int b){return a<b?a:b;}
DEVINL int iabsi(int a){return a<0?-a:a;}

DEVINL double cgc(int j1,int m1,int j2,int m2,int j3,int m3){
  if(m3!=m1+m2) return 0.0;
  if(iabsi(m1)>j1||iabsi(m2)>j2||iabsi(m3)>j3) return 0.0;
  if(j3<iabsi(j1-j2)||j3>j1+j2) return 0.0;
  double pref=(2.0*j3+1.0)*dfact(j3+j1-j2)*dfact(j3-j1+j2)*dfact(j1+j2-j3)/dfact(j1+j2+j3+1);
  pref*=dfact(j3+m3)*dfact(j3-m3)*dfact(j1-m1)*dfact(j1+m1)*dfact(j2-m2)*dfact(j2+m2);
  int kmin=imaxi(0,imaxi(j2-j3-m1,j1-j3+m2));
  int kmax=imini(j1+j2-j3,imini(j1-m1,j2+m2));
  double s=0.0;
  for(int k=kmin;k<=kmax;++k){
    double d=dfact(k)*dfact(j1+j2-j3-k)*dfact(j1-m1-k)*dfact(j2+m2-k)*dfact(j3-j2+m1+k)*dfact(j3-j1-m2+k);
    s += ((k&1)?-1.0:1.0)/d;
  }
  return s*sqrt(pref);
}

// Q[a][b] of e3nn real->complex change of basis, including (-i)^l factor.
DEVINL void qval(int l,int a,int b,double& qr,double& qi){
  int m=a-l; double r=0.0,i=0.0; const double is2=0.70710678118654752440;
  if(m<0){ if(b==l-m) r=is2; else if(b==l+m) i=-is2; }
  else if(m==0){ if(b==l) r=1.0; }
  else { double s=(m&1)?-1.0:1.0; if(b==l+m) r=s*is2; else if(b==l-m) i=s*is2; }
  switch(l&3){
    case 0: qr=r;  qi=i;  break;
    case 1: qr=i;  qi=-r; break;
    case 2: qr=-r; qi=-i; break;
    default:qr=-i; qi=r;  break;
  }
}

__global__ void w3j_kernel(float* __restrict__ c3j){
  const int l1s[13]={1,1,1,1,1,2,2,2,2,2,3,3,3};
  const int l2s[13]={1,1,1,2,2,1,1,2,2,2,1,2,2};
  const int l3s[13]={0,1,2,1,2,1,2,0,1,2,2,1,2};
  const int offs[13]={OFF110,OFF111,OFF112,OFF121,OFF122,OFF211,OFF212,OFF220,OFF221,OFF222,OFF312,OFF321,OFF322};
  int t=blockIdx.x;
  int l1=l1s[t],l2=l2s[t],l3=l3s[t];
  int d1=2*l1+1,d2=2*l2+1,d3=2*l3+1;
  int tot=d1*d2*d3;
  int idx=threadIdx.x;
  if(idx>=tot) return;
  int mo=idx%d3, tmp=idx/d3, lo=tmp%d2, jo=tmp/d2;
  double acc=0.0;
  for(int i=0;i<d1;++i)for(int k=0;k<d2;++k){
    int m1=i-l1,m2=k-l2,m3=m1+m2;
    if(iabsi(m3)>l3) continue;
    double cg=cgc(l1,m1,l2,m2,l3,m3);
    if(cg==0.0) continue;
    int n=l3+m3;
    double q1r,q1i,q2r,q2i,q3r,q3i;
    qval(l1,i,jo,q1r,q1i); qval(l2,k,lo,q2r,q2i); qval(l3,n,mo,q3r,q3i);
    q3i=-q3i; // conj
    double pr=q1r*q2r-q1i*q2i, pi=q1r*q2i+q1i*q2r;
    acc += (pr*q3r - pi*q3i)*cg;
  }
  c3j[offs[t]+idx]=(float)acc;
}

// -------- combined weight: Wc[u,v] = Wl[u,v]/sqrt(U) + (u<Usc ? 1.5*Ws[u,v]/sqrt(Usc) : 0), zero-padded to Vpad cols
__global__ void prep_w_kernel(const float* __restrict__ wl,const float* __restrict__ wsc,
                              float* __restrict__ out,int U,int V,int Usc,int Vpad){
  int i=blockIdx.x*blockDim.x+threadIdx.x;
  if(i>=U*Vpad) return;
  int u=i/Vpad, v=i-u*Vpad;
  float val=0.0f;
  if(v<V){
    val=wl[u*V+v]*(1.0f/sqrtf((float)U));
    if(wsc!=nullptr && u<Usc) val+=1.5f*wsc[u*V+v]*(1.0f/sqrtf((float)Usc));
  }
  out[i]=val;
}

// -------- small contraction helpers (all fully unrolled, register resident) ----------------
template<int I,int J,int K>
DEVINL void contract_sh(const float* sh,const float* C,float* M){
#pragma unroll
  for(int j=0;j<J;++j)
#pragma unroll
    for(int k=0;k<K;++k){
      float v=0.0f;
#pragma unroll
      for(int i=0;i<I;++i) v=fmaf(sh[i],C[(i*J+j)*K+k],v);
      M[j*K+k]=v;
    }
}
template<int J,int K>
DEVINL void apply_f(const float* f,const float* M,float* o){
#pragma unroll
  for(int k=0;k<K;++k){
    float v=0.0f;
#pragma unroll
    for(int j=0;j<J;++j) v=fmaf(f[j],M[j*K+k],v);
    o[k]=v;
  }
}

// -------- edge kernel 1: spherical harmonics + layer-1 aggregation -------------------------
__global__ void edge_sh_kernel(const int* __restrict__ se,const int* __restrict__ re,
    const float* __restrict__ pos,const float* __restrict__ nf,
    float* __restrict__ SH,float* __restrict__ S0,float* __restrict__ Ssh1,
    float* __restrict__ Ssh2,float* __restrict__ indeg,int E){
  int e=blockIdx.x*blockDim.x+threadIdx.x; if(e>=E) return;
  int s=se[e], r=re[e];
  float rx=pos[3*r+0]-pos[3*s+0];
  float ry=pos[3*r+1]-pos[3*s+1];
  float rz=pos[3*r+2]-pos[3*s+2];
  float nrm=sqrtf(rx*rx+ry*ry+rz*rz);
  float inv=1.0f/fmaxf(nrm,1e-12f);
  float x=rx*inv,y=ry*inv,z=rz*inv;
  const float s3=1.7320508075688772f, s10=3.1622776601683795f, s6=2.449489742783178f, s15=3.872983346207417f;
  float y2=y*y, x2z2=x*x+z*z;
  float sh[15];
  sh[0]=x; sh[1]=y; sh[2]=z;
  sh[3]=s3*x*z; sh[4]=s3*x*y; sh[5]=y2-0.5f*x2z2; sh[6]=s3*y*z; sh[7]=0.5f*s3*(z*z-x*x);
  sh[8]=0.25f*s10*x*(3.0f*z*z-x*x);
  sh[9]=s15*x*y*z;
  sh[10]=0.25f*s6*x*(4.0f*y2-x2z2);
  sh[11]=0.5f*y*(2.0f*y2-3.0f*x2z2);
  sh[12]=0.25f*s6*z*(4.0f*y2-x2z2);
  sh[13]=0.5f*s15*y*(z*z-x*x);
  sh[14]=0.25f*s10*z*(z*z-3.0f*x*x);
#pragma unroll
  for(int i=0;i<15;++i) SH[(size_t)e*15+i]=sh[i];
  float a=nf[s];
  unsafeAtomicAdd(&indeg[r],1.0f);
  unsafeAtomicAdd(&S0[r],a);
#pragma unroll
  for(int k=0;k<3;++k) unsafeAtomicAdd(&Ssh1[3*r+k],a*sh[k]);
#pragma unroll
  for(int k=0;k<5;++k) unsafeAtomicAdd(&Ssh2[5*r+k],a*sh[3+k]);
}

// -------- layer-1 node kernel (mul_in = 1 linears, done directly) --------------------------
__global__ void node1_kernel(const float* __restrict__ S0,const float* __restrict__ Ssh1,
    const float* __restrict__ Ssh2,const float* __restrict__ indeg,
    const float* __restrict__ wl0,const float* __restrict__ wsc0,
    const float* __restrict__ wl1,const float* __restrict__ wl2,
    float* __restrict__ cntf,float* __restrict__ F0,float* __restrict__ F1,
    float* __restrict__ F2,int N){
  int n=blockIdx.x*blockDim.x+threadIdx.x; if(n>=N) return;
  float cnt=fmaxf(indeg[n],1.0f); cntf[n]=cnt;
  float inv=1.0f/cnt;
  float ssc=S0[n]*inv;          // shortcut aggregation (no DENOM)
  float alin=ssc*(1.0f/1.5f);   // TP aggregation (with DENOM)
#pragma unroll 8
  for(int v=0;v<32;++v) F0[(size_t)n*32+v]=alin*wl0[v]+ssc*wsc0[v];
  float g=inv*(1.0f/1.5f);
  float t1[3],t2[5];
#pragma unroll
  for(int k=0;k<3;++k) t1[k]=Ssh1[3*n+k]*g;
#pragma unroll
  for(int k=0;k<5;++k) t2[k]=Ssh2[5*n+k]*g;
#pragma unroll
  for(int u=0;u<8;++u){
#pragma unroll
    for(int k=0;k<3;++k) F1[(size_t)n*24+u*3+k]=t1[k]*wl1[u];
#pragma unroll
    for(int k=0;k<5;++k) F2[(size_t)n*40+u*5+k]=t2[k]*wl2[u];
  }
}

// -------- layer-2 edge tensor-product kernel -----------------------------------------------
__global__ void edge2_kernel(const int* __restrict__ se,const int* __restrict__ re,
    const float* __restrict__ SH,const float* __restrict__ F0,
    const float* __restrict__ F1,const float* __restrict__ F2,
    const float* __restrict__ c3j,
    float* __restrict__ A0,float* __restrict__ A1E,float* __restrict__ A1O,
    float* __restrict__ A2E,float* __restrict__ A2O,int E){
  __shared__ float sC[C3J_TOTAL];
  for(int i=threadIdx.x;i<C3J_TOTAL;i+=blockDim.x) sC[i]=c3j[i];
  __syncthreads();
  int e=blockIdx.x*blockDim.x+threadIdx.x;
  if(e>=E) return;
  int s=se[e], r=re[e];
  float sh[15];
#pragma unroll
  for(int i=0;i<15;++i) sh[i]=SH[(size_t)e*15+i];
  const float* sh1=sh; const float* sh2=sh+3; const float* sh3=sh+8;
  float b[24], c[40];
#pragma unroll
  for(int i=0;i<24;++i) b[i]=F1[(size_t)s*24+i];
#pragma unroll
  for(int i=0;i<40;++i) c[i]=F2[(size_t)s*40+i];
  float* pA0 =A0 +(size_t)r*48;
  float* pA1E=A1E+(size_t)r*48;
  float* pA1O=A1O+(size_t)r*192;
  float* pA2E=A2E+(size_t)r*320;
  float* pA2O=A2O+(size_t)r*120;
  float t1[1],t3[3],t5[5];
  // ---- l1 = 1 paths ----
  {
    float M110[3],M111[9],M112[15],M121[15],M122[25];
    contract_sh<3,3,1>(sh1,sC+OFF110,M110);
    contract_sh<3,3,3>(sh1,sC+OFF111,M111);
    contract_sh<3,3,5>(sh1,sC+OFF112,M112);
    contract_sh<3,5,3>(sh1,sC+OFF121,M121);
    contract_sh<3,5,5>(sh1,sC+OFF122,M122);
#pragma unroll
    for(int u=0;u<8;++u){
      const float* bu=b+u*3; const float* cu=c+u*5;
      apply_f<3,1>(bu,M110,t1); unsafeAtomicAdd(pA0+32+u,t1[0]);                 // 1x1->0e rows 32..39
      apply_f<3,3>(bu,M111,t3);
#pragma unroll
      for(int k=0;k<3;++k) unsafeAtomicAdd(pA1E+u*3+k,t3[k]);                    // 1x1->1e rows 0..7
      apply_f<3,5>(bu,M112,t5);
#pragma unroll
      for(int k=0;k<5;++k) unsafeAtomicAdd(pA2E+(8+u)*5+k,t5[k]);                // 1x1->2e rows 8..15
      apply_f<5,3>(cu,M121,t3);
#pragma unroll
      for(int k=0;k<3;++k) unsafeAtomicAdd(pA1O+(40+u)*3+k,t3[k]);               // 1x2->1o rows 40..47
      apply_f<5,5>(cu,M122,t5);
#pragma unroll
      for(int k=0;k<5;++k) unsafeAtomicAdd(pA2O+u*5+k,t5[k]);                    // 1x2->2o rows 0..7
    }
  }
  // ---- l1 = 2 paths ----
  {
    float M211[9],M212[15],M220[5],M221[15],M222[25];
    contract_sh<5,3,3>(sh2,sC+OFF211,M211);
    contract_sh<5,3,5>(sh2,sC+OFF212,M212);
    contract_sh<5,5,1>(sh2,sC+OFF220,M220);
    contract_sh<5,5,3>(sh2,sC+OFF221,M221);
    contract_sh<5,5,5>(sh2,sC+OFF222,M222);
#pragma unroll
    for(int u=0;u<8;++u){
      const float* bu=b+u*3; const float* cu=c+u*5;
      apply_f<3,3>(bu,M211,t3);
#pragma unroll
      for(int k=0;k<3;++k) unsafeAtomicAdd(pA1O+(48+u)*3+k,t3[k]);               // 2x1->1o rows 48..55
      apply_f<3,5>(bu,M212,t5);
#pragma unroll
      for(int k=0;k<5;++k) unsafeAtomicAdd(pA2O+(8+u)*5+k,t5[k]);                // 2x1->2o rows 8..15
      apply_f<5,1>(cu,M220,t1); unsafeAtomicAdd(pA0+40+u,t1[0]);                 // 2x2->0e rows 40..47
      apply_f<5,3>(cu,M221,t3);
#pragma unroll
      for(int k=0;k<3;++k) unsafeAtomicAdd(pA1E+(8+u)*3+k,t3[k]);                // 2x2->1e rows 8..15
      apply_f<5,5>(cu,M222,t5);
#pragma unroll
      for(int k=0;k<5;++k) unsafeAtomicAdd(pA2E+(48+u)*5+k,t5[k]);               // 2x2->2e rows 48..55
    }
  }
  // ---- l1 = 3 paths ----
  {
    float M312[15],M321[15],M322[25];
    contract_sh<7,3,5>(sh3,sC+OFF312,M312);
    contract_sh<7,5,3>(sh3,sC+OFF321,M321);
    contract_sh<7,5,5>(sh3,sC+OFF322,M322);
#pragma unroll
    for(int u=0;u<8;++u){
      const float* bu=b+u*3; const float* cu=c+u*5;
      apply_f<3,5>(bu,M312,t5);
#pragma unroll
      for(int k=0;k<5;++k) unsafeAtomicAdd(pA2E+(56+u)*5+k,t5[k]);               // 3x1->2e rows 56..63
      apply_f<5,3>(cu,M321,t3);
#pragma unroll
      for(int k=0;k<3;++k) unsafeAtomicAdd(pA1O+(56+u)*3+k,t3[k]);               // 3x2->1o rows 56..63
      apply_f<5,5>(cu,M322,t5);
#pragma unroll
      for(int k=0;k<5;++k) unsafeAtomicAdd(pA2O+(16+u)*5+k,t5[k]);               // 3x2->2o rows 16..23
    }
  }
  // ---- sent features (first chunks; also the shortcut source) ----
#pragma unroll
  for(int u=0;u<8;++u){
#pragma unroll
    for(int k=0;k<3;++k) unsafeAtomicAdd(pA1O+u*3+k,b[u*3+k]);
#pragma unroll
    for(int k=0;k<5;++k) unsafeAtomicAdd(pA2E+u*5+k,c[u*5+k]);
  }
  // ---- scalar (0e) sender paths: l x 0e -> l is identity on sh ----
  const float* f0p=F0+(size_t)s*32;
#pragma unroll 4
  for(int u=0;u<32;++u){
    float a=f0p[u];
    unsafeAtomicAdd(pA0+u,a);                                                    // sent 0e rows 0..31
#pragma unroll
    for(int k=0;k<3;++k) unsafeAtomicAdd(pA1O+(8+u)*3+k,a*sh1[k]);               // 1x0e->1o rows 8..39
#pragma unroll
    for(int k=0;k<5;++k) unsafeAtomicAdd(pA2E+(16+u)*5+k,a*sh2[k]);              // 2x0e->2e rows 16..47
  }
}

// -------- WMMA fp32 GEMM: D[(n,k),v] = (A[(n,k),u] * g[n]) x Wc[u,v] ----------------------
// A layout: A[n*U*numM + u*numM + k]; D layout: D[n*V*numM + v*numM + k]; Wc: U x Vpad.
// Two independent accumulator chains (even/odd K-steps) to halve the WMMA D->C RAW chain;
// U is always a multiple of 8.
__global__ void gemm_wmma_kernel(const float* __restrict__ A,const float* __restrict__ W,
    float* __restrict__ D,const float* __restrict__ cnt,
    int M,int U,int V,int Vpad,int numM){
  int lane=threadIdx.x&31;
  int half=lane>>4;
  int l16 =lane&15;
  int row0=blockIdx.x*16;
  int colW=blockIdx.y*16+l16;
  int arow=row0+l16; if(arow>=M) arow=M-1;
  int an=arow/numM;
  int ak=arow-an*numM;
  float g=1.0f/(cnt[an]*1.5f);
  const float* Ap=A+(size_t)an*U*numM+ak;
  const float* Wp=W+colW;
  v8f acc0={}, acc1={};
  for(int k0=0;k0<U;k0+=8){
    int ka=k0+2*half;                     // A 16x4 f32 layout: VGPR v, lane-half h -> K = v + 2h
    v2f a0,b0,a1,b1;
    a0.x=Ap[(size_t)ka*numM]*g;
    a0.y=Ap[(size_t)(ka+1)*numM]*g;
    b0.x=Wp[(size_t)ka*Vpad];             // B 4x16: VGPR v, half h -> K-row = v + 2h, N = lane&15
    b0.y=Wp[(size_t)(ka+1)*Vpad];
    int kb=ka+4;
    a1.x=Ap[(size_t)kb*numM]*g;
    a1.y=Ap[(size_t)(kb+1)*numM]*g;
    b1.x=Wp[(size_t)kb*Vpad];
    b1.y=Wp[(size_t)(kb+1)*Vpad];
    acc0=__builtin_amdgcn_wmma_f32_16x16x4_f32(false,a0,false,b0,(short)0,acc0,false,false);
    acc1=__builtin_amdgcn_wmma_f32_16x16x4_f32(false,a1,false,b1,(short)0,acc1,false,false);
  }
  v8f acc=acc0+acc1;
#pragma unroll
  for(int rr=0;rr<8;++rr){
    int orow=row0+rr+8*half;              // C/D: VGPR r, half h -> M = r + 8h, N = lane&15
    if(orow<M && colW<V){
      int on=orow/numM;
      int ok=orow-on*numM;
      D[(size_t)on*V*numM+(size_t)colW*numM+ok]=acc[rr];
    }
  }
}

// -------- layer-3 edge kernel --------------------------------------------------------------
__global__ void edge3_kernel(const int* __restrict__ se,const int* __restrict__ re,
    const float* __restrict__ SH,
    const float* __restrict__ F0,const float* __restrict__ F1E,const float* __restrict__ F1O,
    const float* __restrict__ F2E,const float* __restrict__ F2O,
    const float* __restrict__ c3j,float* __restrict__ G0E,float* __restrict__ G0O,int E){
  int e=blockIdx.x*blockDim.x+threadIdx.x; if(e>=E) return;
  int s=se[e], r=re[e];
  float sh1[3],sh2[5];
#pragma unroll
  for(int i=0;i<3;++i) sh1[i]=SH[(size_t)e*15+i];
#pragma unroll
  for(int i=0;i<5;++i) sh2[i]=SH[(size_t)e*15+3+i];
  float M110[3],M220[5];
  contract_sh<3,3,1>(sh1,c3j+OFF110,M110);
  contract_sh<5,5,1>(sh2,c3j+OFF220,M220);
  float* g0e=G0E+(size_t)r*48;
  float* g0o=G0O+(size_t)r*16;
  const float* f0p=F0+(size_t)s*32;
#pragma unroll 4
  for(int u=0;u<32;++u) unsafeAtomicAdd(g0e+u,f0p[u]);                           // sent 0e
  float t1[1];
#pragma unroll
  for(int u=0;u<8;++u){
    apply_f<3,1>(F1O+(size_t)s*24+u*3,M110,t1); unsafeAtomicAdd(g0e+32+u,t1[0]); // 1x1o->0e
    apply_f<5,1>(F2E+(size_t)s*40+u*5,M220,t1); unsafeAtomicAdd(g0e+40+u,t1[0]); // 2x2e->0e
    apply_f<3,1>(F1E+(size_t)s*24+u*3,M110,t1); unsafeAtomicAdd(g0o+u,t1[0]);    // 1x1e->0o
    apply_f<5,1>(F2O+(size_t)s*40+u*5,M220,t1); unsafeAtomicAdd(g0o+8+u,t1[0]);  // 2x2o->0o
  }
}

// -------- per-graph reduction + final output -----------------------------------------------
__global__ void reduce_pred_kernel(const float* __restrict__ o0o,const float* __restrict__ o0e,
    const int* __restrict__ batch,float* __restrict__ pred,int N){
  int n=blockIdx.x*blockDim.x+threadIdx.x; if(n>=N) return;
  int g=batch[n];
  unsafeAtomicAdd(&pred[g*8+0],o0o[n]);
#pragma unroll
  for(int j=0;j<7;++j) unsafeAtomicAdd(&pred[g*8+1+j],o0e[(size_t)n*7+j]);
}

__global__ void finalize_kernel(const float* __restrict__ pred,float* __restrict__ out){
  int g=threadIdx.x; if(g>=8) return;
  float p0=pred[g*8+0],p1=pred[g*8+1];
  out[g*8+0]= p0*p1;
  out[g*8+1]=-p0*p1;
#pragma unroll
  for(int j=0;j<6;++j) out[g*8+2+j]=pred[g*8+2+j];
}

// ===========================================================================================
extern "C" void kernel_launch(void* const* d_in, const int* in_sizes, int n_in,
                              void* d_out, int out_size, void* d_ws, size_t ws_size,
                              hipStream_t stream) {
  const float* nf     =(const float*)d_in[0];
  const float* pos    =(const float*)d_in[1];
  const int*   ei     =(const int*)  d_in[2];
  const int*   batch  =(const int*)  d_in[3];
  const float* wl1_0e =(const float*)d_in[4];
  const float* wl1_1o =(const float*)d_in[5];
  const float* wl1_2e =(const float*)d_in[6];
  const float* wsc1_0e=(const float*)d_in[7];
  const float* wl2_0e =(const float*)d_in[8];
  const float* wl2_1e =(const float*)d_in[9];
  const float* wl2_1o =(const float*)d_in[10];
  const float* wl2_2e =(const float*)d_in[11];
  const float* wl2_2o =(const float*)d_in[12];
  const float* wsc2_0e=(const float*)d_in[13];
  const float* wsc2_1o=(const float*)d_in[14];
  const float* wsc2_2e=(const float*)d_in[15];
  const float* wl3_0o =(const float*)d_in[16];
  const float* wl3_0e =(const float*)d_in[17];
  const float* wsc3_0e=(const float*)d_in[18];

  const int N=in_sizes[0];
  const int E=in_sizes[2]/2;
  const int* se=ei;
  const int* re=ei+E;

  float* ws=(float*)d_ws;
  size_t off=0;
  auto take=[&](size_t n){ size_t r=off; off+=n; return r; };

  float* C3J   =ws+take(1024);
  float* WC2_0E=ws+take(48*32);
  float* WC2_1E=ws+take(16*16);
  float* WC2_1O=ws+take(64*16);
  float* WC2_2E=ws+take(64*16);
  float* WC2_2O=ws+take(24*16);
  float* WC3_0O=ws+take(16*16);
  float* WC3_0E=ws+take(48*16);
  float* SH    =ws+take((size_t)E*15);
  float* CNT   =ws+take(N);
  float* F1_0E =ws+take((size_t)N*32);
  float* F1_1O =ws+take((size_t)N*24);
  float* F1_2E =ws+take((size_t)N*40);
  float* F2_0E =ws+take((size_t)N*32);
  float* F2_1E =ws+take((size_t)N*24);
  float* F2_1O =ws+take((size_t)N*24);
  float* F2_2E =ws+take((size_t)N*40);
  float* F2_2O =ws+take((size_t)N*40);
  float* O3_0O =ws+take(N);
  float* O3_0E =ws+take((size_t)N*7);
  size_t zb=off;                       // zero-initialized accumulator block
  float* INDEG =ws+take(N);
  float* S0    =ws+take(N);
  float* SSH1  =ws+take((size_t)N*3);
  float* SSH2  =ws+take((size_t)N*5);
  float* A0    =ws+take((size_t)N*48);
  float* A1E   =ws+take((size_t)N*48);
  float* A1O   =ws+take((size_t)N*192);
  float* A2E   =ws+take((size_t)N*320);
  float* A2O   =ws+take((size_t)N*120);
  float* G0E   =ws+take((size_t)N*48);
  float* G0O   =ws+take((size_t)N*16);
  float* PRED  =ws+take(64);
  size_t ze=off;

  hipMemsetAsync(ws+zb,0,(ze-zb)*sizeof(float),stream);

  // constants
  w3j_kernel<<<13,256,0,stream>>>(C3J);
  prep_w_kernel<<<(48*32+255)/256,256,0,stream>>>(wl2_0e,wsc2_0e,WC2_0E,48,32,32,32);
  prep_w_kernel<<<1,256,0,stream>>>(wl2_1e,nullptr,WC2_1E,16,8,0,16);
  prep_w_kernel<<<(64*16+255)/256,256,0,stream>>>(wl2_1o,wsc2_1o,WC2_1O,64,8,8,16);
  prep_w_kernel<<<(64*16+255)/256,256,0,stream>>>(wl2_2e,wsc2_2e,WC2_2E,64,8,8,16);
  prep_w_kernel<<<(24*16+255)/256,256,0,stream>>>(wl2_2o,nullptr,WC2_2O,24,8,0,16);
  prep_w_kernel<<<1,256,0,stream>>>(wl3_0o,nullptr,WC3_0O,16,1,0,16);
  prep_w_kernel<<<(48*16+255)/256,256,0,stream>>>(wl3_0e,wsc3_0e,WC3_0E,48,7,32,16);

  // layer 1
  edge_sh_kernel<<<(E+255)/256,256,0,stream>>>(se,re,pos,nf,SH,S0,SSH1,SSH2,INDEG,E);
  node1_kernel<<<(N+127)/128,128,0,stream>>>(S0,SSH1,SSH2,INDEG,wl1_0e,wsc1_0e,wl1_1o,wl1_2e,
                                             CNT,F1_0E,F1_1O,F1_2E,N);
  // layer 2
  edge2_kernel<<<(E+127)/128,128,0,stream>>>(se,re,SH,F1_0E,F1_1O,F1_2E,C3J,A0,A1E,A1O,A2E,A2O,E);
  dim3 blk(32,1,1);
  gemm_wmma_kernel<<<dim3((N*1+15)/16,2),blk,0,stream>>>(A0 ,WC2_0E,F2_0E,CNT,N*1,48,32,32,1);
  gemm_wmma_kernel<<<dim3((N*3+15)/16,1),blk,0,stream>>>(A1E,WC2_1E,F2_1E,CNT,N*3,16, 8,16,3);
  gemm_wmma_kernel<<<dim3((N*3+15)/16,1),blk,0,stream>>>(A1O,WC2_1O,F2_1O,CNT,N*3,64, 8,16,3);
  gemm_wmma_kernel<<<dim3((N*5+15)/16,1),blk,0,stream>>>(A2E,WC2_2E,F2_2E,CNT,N*5,64, 8,16,5);
  gemm_wmma_kernel<<<dim3((N*5+15)/16,1),blk,0,stream>>>(A2O,WC2_2O,F2_2O,CNT,N*5,24, 8,16,5);
  // layer 3
  edge3_kernel<<<(E+255)/256,256,0,stream>>>(se,re,SH,F2_0E,F2_1E,F2_1O,F2_2E,F2_2O,C3J,G0E,G0O,E);
  gemm_wmma_kernel<<<dim3((N+15)/16,1),blk,0,stream>>>(G0O,WC3_0O,O3_0O,CNT,N,16,1,16,1);
  gemm_wmma_kernel<<<dim3((N+15)/16,1),blk,0,stream>>>(G0E,WC3_0E,O3_0E,CNT,N,48,7,16,1);
  // readout
  reduce_pred_kernel<<<(N+255)/256,256,0,stream>>>(O3_0O,O3_0E,batch,PRED,N);
  finalize_kernel<<<1,8,0,stream>>>(PRED,(float*)d_out);
}